// MolecularGraphNeuralNetwork_9156870275185
// MI455X (gfx1250) — compile-verified
//
#include <hip/hip_runtime.h>
#include <hip/hip_bf16.h>
#include <math.h>

// ---------------------------------------------------------------------------
// MolecularGraphNeuralNetwork on MI455X (gfx1250)
//
// dist is block-diagonal with pad 1e6 -> exp(-g*1e12) == 0.0f in fp32, so
// M @ hv reduces to 256 independent 32x32 block multiplies.  The heavy
// compute is twelve [*,200/204] x [200/204,*] fp32 GEMMs, run on
// V_WMMA_F32_16X16X4_F32 with 2x2 register blocking: each wave owns a 32x32
// C macro-tile, issuing 4 WMMAs per 2 A-frag + 2 B-frag loads.  N is padded
// to 224 (14 tiles = 7 tile pairs) so the macro grid is exact and EXEC stays
// all-ones (WMMA requirement).  Working set (~18 MB) is L2-resident.
// ---------------------------------------------------------------------------

typedef float v2f __attribute__((ext_vector_type(2)));
typedef float v8f __attribute__((ext_vector_type(8)));

#define NATOMS_TOT 8192
#define DIM        200
#define DIMP       224      // padded N (14 tiles of 16 -> 7 tile-pairs)
#define PREDK      204      // DIM + NADD
#define NMOL       256
#define NA         32
#define NVOCAB     100

// ---------------- gather: av[i,d] = embed[atoms[i], d] ----------------------
__global__ void k_gather(const int* __restrict__ atoms,
                         const float* __restrict__ emb,
                         float* __restrict__ AV) {
    int tid = blockIdx.x * 256 + threadIdx.x;
    if (tid >= NATOMS_TOT * DIM) return;
    int i = tid / DIM, d = tid - i * DIM;
    AV[(size_t)i * DIMP + d] = emb[atoms[i] * DIM + d];
}

// ------------- extract intra-molecule d^2 blocks [256][32][32] --------------
__global__ void k_d2block(const float* __restrict__ dist,
                          float* __restrict__ d2b) {
    int tid = blockIdx.x * 256 + threadIdx.x;
    if (tid >= NMOL * NA * NA) return;
    int b = tid >> 10;
    int e = tid & 1023;
    int i = e >> 5, j = e & 31;
    float d = dist[(size_t)(b * NA + i) * NATOMS_TOT + (b * NA + j)];
    d2b[tid] = d * d;
}

// ---------------- pad K x Nsrc matrix into K x Ndst (zero fill) -------------
__global__ void k_pad(const float* __restrict__ src, float* __restrict__ dst,
                      int K, int Nsrc, int Ndst) {
    int tid = blockIdx.x * 256 + threadIdx.x;
    if (tid >= K * Ndst) return;
    int r = tid / Ndst, c = tid - r * Ndst;
    dst[tid] = (c < Nsrc) ? src[r * Nsrc + c] : 0.0f;
}

// ------------- WMMA fp32 GEMM, 2x2 blocked: C = [relu](A @ B + bias) --------
// A: Mrows x K (lda), B: K x N (ldb, N padded to 32), C: Mrows x N (ldc).
// One wave computes a 32x32 macro-tile (4 WMMA accumulators).
__global__ void k_gemm_wmma(const float* __restrict__ A,
                            const float* __restrict__ B,
                            const float* __restrict__ bias,
                            float* __restrict__ C,
                            int Mrows, int K, int lda, int ldb, int ldc,
                            int npairs, int do_relu) {
    int wave = threadIdx.x >> 5;
    int np = blockIdx.y * 4 + wave;
    if (np >= npairs) return;               // wave-uniform: EXEC stays all-1s
    int mp = blockIdx.x;

    int lane = threadIdx.x & 31;
    int half = lane >> 4;                   // selects K-pair within 16x4 frag
    int l16  = lane & 15;

    int m0 = mp * 32;
    int n0 = np * 32;

    const float* arow0 = A + (size_t)(m0 + l16) * lda + 2 * half;
    const float* arow1 = arow0 + (size_t)16 * lda;
    const float* bptr0 = B + (size_t)(2 * half) * ldb + n0 + l16;
    const float* bptr1 = bptr0 + 16;

    v8f c00 = {}, c01 = {}, c10 = {}, c11 = {};
    for (int k0 = 0; k0 < K; k0 += 4) {
        v2f a0, a1, b0, b1;
        a0.x = arow0[k0];
        a0.y = arow0[k0 + 1];
        a1.x = arow1[k0];
        a1.y = arow1[k0 + 1];
        b0.x = bptr0[(size_t)k0 * ldb];
        b0.y = bptr0[(size_t)(k0 + 1) * ldb];
        b1.x = bptr1[(size_t)k0 * ldb];
        b1.y = bptr1[(size_t)(k0 + 1) * ldb];
        c00 = __builtin_amdgcn_wmma_f32_16x16x4_f32(false, a0, false, b0,
                                                    (short)0, c00, false, false);
        c01 = __builtin_amdgcn_wmma_f32_16x16x4_f32(false, a0, false, b1,
                                                    (short)0, c01, false, false);
        c10 = __builtin_amdgcn_wmma_f32_16x16x4_f32(false, a1, false, b0,
                                                    (short)0, c10, false, false);
        c11 = __builtin_amdgcn_wmma_f32_16x16x4_f32(false, a1, false, b1,
                                                    (short)0, c11, false, false);
    }

    int col0 = n0 + l16;
    int col1 = col0 + 16;
    float bv0 = bias ? bias[col0] : 0.0f;
    float bv1 = bias ? bias[col1] : 0.0f;
#pragma unroll
    for (int v = 0; v < 8; ++v) {
        int row0 = m0 + v + 8 * half;
        int row1 = row0 + 16;
        float v00 = c00[v] + bv0;
        float v01 = c01[v] + bv1;
        float v10 = c10[v] + bv0;
        float v11 = c11[v] + bv1;
        if (do_relu) {
            v00 = fmaxf(v00, 0.0f); v01 = fmaxf(v01, 0.0f);
            v10 = fmaxf(v10, 0.0f); v11 = fmaxf(v11, 0.0f);
        }
        C[(size_t)row0 * ldc + col0] = v00;
        C[(size_t)row0 * ldc + col1] = v01;
        C[(size_t)row1 * ldc + col0] = v10;
        C[(size_t)row1 * ldc + col1] = v11;
    }
}

// ---- per-molecule: av = normalize(hv + M @ hv), M = exp(-g[atom_j]*d2) -----
// One workgroup (256 threads) per molecule. 8 threads per row, 25 cols each.
__global__ void k_combine_norm(const float* __restrict__ HV,
                               const float* __restrict__ d2b,
                               const int* __restrict__ atoms,
                               const float* __restrict__ gamma_row,
                               float* __restrict__ AV) {
    __shared__ float shv[NA * DIM];     // 25.6 KB
    __shared__ float shm[NA * NA];      // 4 KB
    __shared__ float spart[NA * 8];

    int b = blockIdx.x;
    int t = threadIdx.x;

    for (int e = t; e < NA * DIM; e += 256) {
        int i = e / DIM, d = e - i * DIM;
        shv[e] = HV[(size_t)(b * NA + i) * DIMP + d];
    }
    for (int e = t; e < NA * NA; e += 256) {
        int j = e & 31;
        float g = gamma_row[atoms[b * NA + j]];
        shm[e] = expf(-g * d2b[b * NA * NA + e]);
    }
    __syncthreads();

    int i  = t >> 3;          // row 0..31
    int c0 = (t & 7) * 25;    // col start (stride-25 avoids bank conflicts)

    float out[25];
    float ss = 0.0f;
#pragma unroll
    for (int dd = 0; dd < 25; ++dd) {
        int d = c0 + dd;
        float o = shv[i * DIM + d];
#pragma unroll
        for (int j = 0; j < NA; ++j)
            o = fmaf(shm[i * NA + j], shv[j * DIM + d], o);
        out[dd] = o;
        ss += o * o;
    }
    spart[t] = ss;
    __syncthreads();

    ss = 0.0f;
#pragma unroll
    for (int p = 0; p < 8; ++p) ss += spart[i * 8 + p];
    float inv = 1.0f / fmaxf(sqrtf(ss), 1e-12f);

#pragma unroll
    for (int dd = 0; dd < 25; ++dd)
        AV[(size_t)(b * NA + i) * DIMP + c0 + dd] = out[dd] * inv;
}

// ------------- per-molecule sum over atoms + adduct concat ------------------
__global__ void k_molsum(const float* __restrict__ AV,
                         const float* __restrict__ adducts,
                         float* __restrict__ MV) {
    int tid = blockIdx.x * 256 + threadIdx.x;
    if (tid >= NMOL * DIMP) return;
    int b = tid / DIMP, d = tid - b * DIMP;
    float v = 0.0f;
    if (d < DIM) {
        for (int i = 0; i < NA; ++i)
            v += AV[(size_t)(b * NA + i) * DIMP + d];
    } else if (d < PREDK) {
        v = adducts[b * 4 + (d - DIM)];
    }
    MV[tid] = v;
}

// ---------------- final: out[b] = mv[b] . W_prop + b_prop -------------------
__global__ void k_final(const float* __restrict__ MV,
                        const float* __restrict__ Wp,
                        const float* __restrict__ bp,
                        float* __restrict__ out) {
    int b = blockIdx.x;
    int l = threadIdx.x;               // 32 threads (one wave)
    float s = 0.0f;
    for (int d = l; d < PREDK; d += 32)
        s += MV[(size_t)b * DIMP + d] * Wp[d];
#pragma unroll
    for (int off = 16; off > 0; off >>= 1)
        s += __shfl_down(s, off, 32);
    if (l == 0) out[b] = s + bp[0];
}

// ---------------------------------------------------------------------------
extern "C" void kernel_launch(void* const* d_in, const int* in_sizes, int n_in,
                              void* d_out, int out_size, void* d_ws, size_t ws_size,
                              hipStream_t stream) {
    const int*   atoms   = (const int*)  d_in[0];
    const float* dist    = (const float*)d_in[1];
    const float* adducts = (const float*)d_in[2];
    const float* emb     = (const float*)d_in[3];
    const float* gamma   = (const float*)d_in[4];
    const float* Waw     = (const float*)d_in[5];   // [6,200,200]
    const float* Wab     = (const float*)d_in[6];   // [6,200]
    const float* Wow     = (const float*)d_in[7];   // [3,200,200]
    const float* Wob     = (const float*)d_in[8];   // [3,200]
    const float* Wpw     = (const float*)d_in[9];   // [3,204,204]
    const float* Wpb     = (const float*)d_in[10];  // [3,204]
    const float* Wprop   = (const float*)d_in[11];  // [204,1]
    const float* bprop   = (const float*)d_in[12];  // [1]
    float* out = (float*)d_out;

    // workspace layout (floats)
    float* ws  = (float*)d_ws;
    const size_t SZ_ACT = (size_t)NATOMS_TOT * DIMP;        // 1,835,008
    float* AV  = ws;
    float* HV  = AV + SZ_ACT;
    float* D2B = HV + SZ_ACT;                               // 262,144
    float* WA  = D2B + (size_t)NMOL * NA * NA;              // 6 x 200x224
    float* BA  = WA + 6 * (size_t)DIM * DIMP;               // 6 x 224
    float* WO  = BA + 6 * DIMP;                             // 3 x 200x224
    float* BO  = WO + 3 * (size_t)DIM * DIMP;               // 3 x 224
    float* WP  = BO + 3 * DIMP;                             // 3 x 204x224
    float* BP  = WP + 3 * (size_t)PREDK * DIMP;             // 3 x 224
    float* MV  = BP + 3 * DIMP;                             // 256x224
    float* MV2 = MV + (size_t)NMOL * DIMP;

    const dim3 blk256(256);

    // ---- prep: gather embeddings, extract d2 blocks, pad weights/biases ----
    k_gather<<<dim3((NATOMS_TOT * DIM + 255) / 256), blk256, 0, stream>>>(atoms, emb, AV);
    k_d2block<<<dim3((NMOL * NA * NA + 255) / 256), blk256, 0, stream>>>(dist, D2B);

    for (int l = 0; l < 6; ++l) {
        k_pad<<<dim3((DIM * DIMP + 255) / 256), blk256, 0, stream>>>(
            Waw + (size_t)l * DIM * DIM, WA + (size_t)l * DIM * DIMP, DIM, DIM, DIMP);
        k_pad<<<dim3((DIMP + 255) / 256), blk256, 0, stream>>>(
            Wab + (size_t)l * DIM, BA + (size_t)l * DIMP, 1, DIM, DIMP);
    }
    for (int l = 0; l < 3; ++l) {
        k_pad<<<dim3((DIM * DIMP + 255) / 256), blk256, 0, stream>>>(
            Wow + (size_t)l * DIM * DIM, WO + (size_t)l * DIM * DIMP, DIM, DIM, DIMP);
        k_pad<<<dim3((DIMP + 255) / 256), blk256, 0, stream>>>(
            Wob + (size_t)l * DIM, BO + (size_t)l * DIMP, 1, DIM, DIMP);
        k_pad<<<dim3((PREDK * DIMP + 255) / 256), blk256, 0, stream>>>(
            Wpw + (size_t)l * PREDK * PREDK, WP + (size_t)l * PREDK * DIMP, PREDK, PREDK, DIMP);
        k_pad<<<dim3((DIMP + 255) / 256), blk256, 0, stream>>>(
            Wpb + (size_t)l * PREDK, BP + (size_t)l * DIMP, 1, PREDK, DIMP);
    }

    const int NPAIRS = DIMP / 32;                   // 7 macro-cols
    const dim3 gblk(128);                           // 4 waves / block
    const dim3 ggrid_big(NATOMS_TOT / 32, (NPAIRS + 3) / 4);
    const dim3 ggrid_sml(NMOL / 32,       (NPAIRS + 3) / 4);

    // ---- 6 hidden layers: hv = relu(av@W+b); av = normalize(hv + M@hv) -----
    for (int l = 0; l < 6; ++l) {
        k_gemm_wmma<<<ggrid_big, gblk, 0, stream>>>(
            AV, WA + (size_t)l * DIM * DIMP, BA + (size_t)l * DIMP, HV,
            NATOMS_TOT, DIM, DIMP, DIMP, DIMP, NPAIRS, 1);
        k_combine_norm<<<dim3(NMOL), blk256, 0, stream>>>(
            HV, D2B, atoms, gamma + (size_t)l * NVOCAB, AV);
    }

    // ---- 3 output layers: av = relu(av@W+b), ping-pong AV<->HV -------------
    float* pa = AV; float* pb = HV;
    for (int l = 0; l < 3; ++l) {
        k_gemm_wmma<<<ggrid_big, gblk, 0, stream>>>(
            pa, WO + (size_t)l * DIM * DIMP, BO + (size_t)l * DIMP, pb,
            NATOMS_TOT, DIM, DIMP, DIMP, DIMP, NPAIRS, 1);
        float* t = pa; pa = pb; pb = t;
    }
    // result of out-layers is in pa

    // ---- molecule sum + adduct concat -> MV [256 x 224] --------------------
    k_molsum<<<dim3((NMOL * DIMP + 255) / 256), blk256, 0, stream>>>(pa, adducts, MV);

    // ---- 3 pred layers (K = 204), ping-pong MV<->MV2 -----------------------
    float* qa = MV; float* qb = MV2;
    for (int l = 0; l < 3; ++l) {
        k_gemm_wmma<<<ggrid_sml, gblk, 0, stream>>>(
            qa, WP + (size_t)l * PREDK * DIMP, BP + (size_t)l * DIMP, qb,
            NMOL, PREDK, DIMP, DIMP, DIMP, NPAIRS, 1);
        float* t = qa; qa = qb; qb = t;
    }
    // result in qa

    // ---- final projection ---------------------------------------------------
    k_final<<<dim3(NMOL), dim3(32), 0, stream>>>(qa, Wprop, bprop, out);
}